// AutoEncoderTopK_12249246728720
// MI455X (gfx1250) — compile-verified
//
#include <hip/hip_runtime.h>
#include <hip/hip_bf16.h>

typedef float v2f __attribute__((ext_vector_type(2)));
typedef float v8f __attribute__((ext_vector_type(8)));

#define ACT_DIM 768
#define DICT 16384
#define TK 64
#define BATCH 8
#define SEQ 256
#define M_TOKENS (BATCH * SEQ)   // 2048
#define NWIN 127                 // (256-4)/2 + 1
#define WSTRIDE 2
#define MASK_WORDS (DICT / 32)   // 512

// ---------------------------------------------------------------------------
// Kernel 1: z = relu((x - b_dec) @ W_enc^T + b_enc), fp32 WMMA 16x16x4.
// A is 16x4 (M x K), B is 4x16 (K x N) with B[k][n] = W_enc[n*768 + k]
// (W_enc row n is contiguous, so each lane reads a float2 from its row).
// Per-wave: one 16(M) x 64(N) strip = 4 accumulators sharing one A fragment.
// Block: 8 waves covering 8 M-tiles. Grid: (DICT/64, M/128) = (256, 16).
// ---------------------------------------------------------------------------
__global__ __launch_bounds__(256) void enc_gemm_kernel(
    const float* __restrict__ x, const float* __restrict__ Wenc,
    const float* __restrict__ b_enc, const float* __restrict__ b_dec,
    float* __restrict__ z)
{
  const int lane = threadIdx.x & 31;
  const int wave = threadIdx.x >> 5;
  const int half = lane >> 4;   // lanes 0-15 -> K=0,1 ; lanes 16-31 -> K=2,3
  const int l    = lane & 15;

  const int m0 = (blockIdx.y * 8 + wave) * 16;
  const int n0 = blockIdx.x * 64;

  const float* Arow = x    + (size_t)(m0 + l) * ACT_DIM;
  const float* B0   = Wenc + (size_t)(n0 +  0 + l) * ACT_DIM;
  const float* B1   = Wenc + (size_t)(n0 + 16 + l) * ACT_DIM;
  const float* B2   = Wenc + (size_t)(n0 + 32 + l) * ACT_DIM;
  const float* B3   = Wenc + (size_t)(n0 + 48 + l) * ACT_DIM;

  v8f acc0 = {}, acc1 = {}, acc2 = {}, acc3 = {};

  for (int k = 0; k < ACT_DIM; k += 4) {
    const int ko = k + 2 * half;
    const float bd0 = b_dec[ko], bd1 = b_dec[ko + 1];
    v2f a;  a.x  = Arow[ko] - bd0; a.y  = Arow[ko + 1] - bd1;
    v2f b0; b0.x = B0[ko];         b0.y = B0[ko + 1];
    v2f b1; b1.x = B1[ko];         b1.y = B1[ko + 1];
    v2f b2; b2.x = B2[ko];         b2.y = B2[ko + 1];
    v2f b3; b3.x = B3[ko];         b3.y = B3[ko + 1];
    acc0 = __builtin_amdgcn_wmma_f32_16x16x4_f32(false, a, false, b0, (short)0, acc0, false, false);
    acc1 = __builtin_amdgcn_wmma_f32_16x16x4_f32(false, a, false, b1, (short)0, acc1, false, false);
    acc2 = __builtin_amdgcn_wmma_f32_16x16x4_f32(false, a, false, b2, (short)0, acc2, false, false);
    acc3 = __builtin_amdgcn_wmma_f32_16x16x4_f32(false, a, false, b3, (short)0, acc3, false, false);
  }

  // C/D layout: VGPR v -> (M = v + 8*half, N = lane&15)
  for (int v = 0; v < 8; ++v) {
    const int row = m0 + v + 8 * half;
    const size_t base = (size_t)row * DICT;
    const int c0 = n0 + l;
    z[base + c0 +  0] = fmaxf(acc0[v] + b_enc[c0 +  0], 0.f);
    z[base + c0 + 16] = fmaxf(acc1[v] + b_enc[c0 + 16], 0.f);
    z[base + c0 + 32] = fmaxf(acc2[v] + b_enc[c0 + 32], 0.f);
    z[base + c0 + 48] = fmaxf(acc3[v] + b_enc[c0 + 48], 0.f);
  }
}

// ---------------------------------------------------------------------------
// Radix select: exact value of the k-th largest among n non-negative floats
// stored in LDS (non-negative IEEE floats order identically as uint32).
// 4 passes of 8-bit digits, MSB first. Returns k-th value; *kRemainOut gets
// (k - count(strictly greater)), i.e. how many ties at the threshold to take.
// ---------------------------------------------------------------------------
__device__ __forceinline__ float radix_select_topk(
    const float* vals, int n, unsigned k,
    unsigned* hist, unsigned* scal, unsigned* kRemainOut)
{
  unsigned prefix = 0;
  unsigned kremain = k;
  for (int pass = 0; pass < 4; ++pass) {
    const int shift = 24 - pass * 8;
    hist[threadIdx.x] = 0;
    __syncthreads();
    const unsigned pmask = (pass == 0) ? 0u : (0xFFFFFFFFu << (shift + 8));
    for (int d = threadIdx.x; d < n; d += 256) {
      const unsigned bits = __float_as_uint(vals[d]);
      if ((bits & pmask) == prefix)
        atomicAdd(&hist[(bits >> shift) & 255u], 1u);
    }
    __syncthreads();
    if (threadIdx.x == 0) {
      unsigned cum = 0, digit = 0;
      for (int bin = 255; bin >= 0; --bin) {
        const unsigned c = hist[bin];
        if (cum + c >= kremain) { digit = (unsigned)bin; break; }
        cum += c;
      }
      kremain -= cum;
      prefix |= digit << shift;
      scal[0] = prefix;
      scal[1] = kremain;
    }
    __syncthreads();
    prefix  = scal[0];
    kremain = scal[1];
  }
  if (kRemainOut) *kRemainOut = kremain;
  return __uint_as_float(prefix);
}

// ---------------------------------------------------------------------------
// Kernel 2: per (batch, window): wsum = sum of 4 z rows; top-64 of wsum;
// emit 16384-bit selection mask.
// ---------------------------------------------------------------------------
__global__ __launch_bounds__(256) void window_topk_kernel(
    const float* __restrict__ z, unsigned* __restrict__ maskW)
{
  __shared__ float    wsum[DICT];
  __shared__ unsigned hist[256];
  __shared__ unsigned scal[2];

  const int bw = blockIdx.x;          // 0 .. BATCH*NWIN-1
  const int b  = bw / NWIN;
  const int w  = bw - b * NWIN;
  const float* zb = z + ((size_t)b * SEQ + (size_t)w * WSTRIDE) * DICT;

  for (int d = threadIdx.x; d < DICT; d += 256)
    wsum[d] = zb[d] + zb[DICT + d] + zb[2 * DICT + d] + zb[3 * DICT + d];
  __syncthreads();

  const float thr = radix_select_topk(wsum, DICT, TK, hist, scal, nullptr);

  unsigned* mw = maskW + (size_t)bw * MASK_WORDS;
  for (int wd = threadIdx.x; wd < MASK_WORDS; wd += 256) {
    unsigned bits = 0;
    const int dbase = wd * 32;
    for (int j = 0; j < 32; ++j)
      bits |= (wsum[dbase + j] >= thr) ? (1u << j) : 0u;
    mw[wd] = bits;
  }
}

// ---------------------------------------------------------------------------
// Kernel 3: per token: fv = z * votecount (from <=2 covering windows),
// exact top-64 of fv (ties broken by smallest index, like jax top_k),
// sparsify z in place, emit compact (idx, val) list for the decoder.
// ---------------------------------------------------------------------------
__global__ __launch_bounds__(256) void token_topk_kernel(
    float* __restrict__ z, const unsigned* __restrict__ maskW,
    int* __restrict__ nnzIdx, float* __restrict__ nnzVal)
{
  __shared__ float    fv[DICT];
  __shared__ unsigned hist[256];
  __shared__ unsigned scal[2];
  __shared__ unsigned tieScan[257];
  __shared__ unsigned appendCnt;

  const int token = blockIdx.x;       // 0 .. 2047
  const int b = token >> 8;
  const int t = token & 255;
  float* zrow = z + (size_t)token * DICT;

  // windows covering t: w*2 <= t <= w*2+3  ->  w in [ceil((t-3)/2), floor(t/2)]
  int wlo = (t - 2) >> 1; if (wlo < 0) wlo = 0;
  int whi = t >> 1;       if (whi > NWIN - 1) whi = NWIN - 1;
  const unsigned* m0p = maskW + (size_t)(b * NWIN + wlo) * MASK_WORDS;
  const unsigned* m1p = maskW + (size_t)(b * NWIN + whi) * MASK_WORDS;
  const bool two = (whi > wlo);

  for (int d = threadIdx.x; d < DICT; d += 256) {
    const int wd = d >> 5;
    const unsigned bit = 1u << (d & 31);
    const float cnt = ((m0p[wd] & bit) ? 1.f : 0.f) +
                      ((two && (m1p[wd] & bit)) ? 1.f : 0.f);
    fv[d] = zrow[d] * cnt;
  }
  if (threadIdx.x == 0) appendCnt = 0;
  if (threadIdx.x < TK) {             // init compact list (ws is poisoned)
    nnzIdx[(size_t)token * TK + threadIdx.x] = 0;
    nnzVal[(size_t)token * TK + threadIdx.x] = 0.f;
  }
  __syncthreads();

  unsigned Tties;
  const float thr = radix_select_topk(fv, DICT, TK, hist, scal, &Tties);

  // Tie ranking in increasing-index order via contiguous per-thread chunks.
  const int chunk = DICT / 256;       // 64
  const int d0 = threadIdx.x * chunk;
  unsigned eqLocal = 0;
  for (int j = 0; j < chunk; ++j) eqLocal += (fv[d0 + j] == thr) ? 1u : 0u;
  tieScan[threadIdx.x] = eqLocal;
  __syncthreads();
  if (threadIdx.x == 0) {
    unsigned run = 0;
    for (int i = 0; i < 256; ++i) { const unsigned c = tieScan[i]; tieScan[i] = run; run += c; }
    tieScan[256] = run;
  }
  __syncthreads();

  unsigned tieRank = tieScan[threadIdx.x];
  for (int j = 0; j < chunk; ++j) {
    const int d = d0 + j;
    const float f = fv[d];
    bool sel = false;
    if (f > thr) sel = true;
    else if (f == thr) { if (tieRank < Tties) sel = true; ++tieRank; }
    const float zv = zrow[d];
    if (sel) {
      const unsigned slot = atomicAdd(&appendCnt, 1u);
      if (slot < TK) {
        nnzIdx[(size_t)token * TK + slot] = d;
        nnzVal[(size_t)token * TK + slot] = zv;
      }
    }
    zrow[d] = sel ? zv : 0.f;
  }
}

// ---------------------------------------------------------------------------
// Kernel 4: sparse decode. recon[token,:] = sum_j val_j * W_enc[idx_j,:] + b_dec
// (W_enc == W_dec^T, so dict atom idx is a contiguous 768-float row.)
// ---------------------------------------------------------------------------
__global__ __launch_bounds__(256) void decode_kernel(
    const int* __restrict__ nnzIdx, const float* __restrict__ nnzVal,
    const float* __restrict__ Wenc, const float* __restrict__ b_dec,
    float* __restrict__ recon)
{
  __shared__ int   sIdx[TK];
  __shared__ float sVal[TK];
  const int token = blockIdx.x;
  if (threadIdx.x < TK) {
    sIdx[threadIdx.x] = nnzIdx[(size_t)token * TK + threadIdx.x];
    sVal[threadIdx.x] = nnzVal[(size_t)token * TK + threadIdx.x];
  }
  __syncthreads();

  const int dA = threadIdx.x, dB = threadIdx.x + 256, dC = threadIdx.x + 512;
  float a0 = 0.f, a1 = 0.f, a2 = 0.f;
  for (int j = 0; j < TK; ++j) {
    const float v = sVal[j];
    const float* row = Wenc + (size_t)sIdx[j] * ACT_DIM;
    a0 += v * row[dA];
    a1 += v * row[dB];
    a2 += v * row[dC];
  }
  float* out = recon + (size_t)token * ACT_DIM;
  out[dA] = a0 + b_dec[dA];
  out[dB] = a1 + b_dec[dB];
  out[dC] = a2 + b_dec[dC];
}

// ---------------------------------------------------------------------------
extern "C" void kernel_launch(void* const* d_in, const int* in_sizes, int n_in,
                              void* d_out, int out_size, void* d_ws, size_t ws_size,
                              hipStream_t stream)
{
  const float* x    = (const float*)d_in[0];
  const float* Wenc = (const float*)d_in[1];
  const float* benc = (const float*)d_in[2];
  const float* Wdec = (const float*)d_in[3];
  const float* bdec = (const float*)d_in[4];
  (void)Wdec; (void)in_sizes; (void)n_in; (void)out_size; (void)ws_size;

  float* recon   = (float*)d_out;                      // 2048*768
  float* encoded = recon + (size_t)M_TOKENS * ACT_DIM; // 2048*16384 (also used as z)

  unsigned char* ws = (unsigned char*)d_ws;
  const size_t maskBytes = (size_t)BATCH * NWIN * MASK_WORDS * sizeof(unsigned); // ~2 MB
  unsigned* maskW = (unsigned*)ws;
  int*   nnzIdx = (int*)(ws + ((maskBytes + 255) & ~(size_t)255));
  float* nnzVal = (float*)((unsigned char*)nnzIdx + (size_t)M_TOKENS * TK * sizeof(int));

  dim3 gGemm(DICT / 64, M_TOKENS / (16 * 8));          // (256, 16)
  enc_gemm_kernel<<<gGemm, 256, 0, stream>>>(x, Wenc, benc, bdec, encoded);
  window_topk_kernel<<<BATCH * NWIN, 256, 0, stream>>>(encoded, maskW);
  token_topk_kernel<<<M_TOKENS, 256, 0, stream>>>(encoded, maskW, nnzIdx, nnzVal);
  decode_kernel<<<M_TOKENS, 256, 0, stream>>>(nnzIdx, nnzVal, Wenc, bdec, recon);
}